// EFGCNLayer_33260226740307
// MI455X (gfx1250) — compile-verified
//
#include <hip/hip_runtime.h>

typedef __attribute__((ext_vector_type(2))) float v2f;
typedef __attribute__((ext_vector_type(8))) float v8f;

static constexpr int BB = 8;     // batch
static constexpr int NN = 2048;  // nodes
static constexpr int FF = 64;    // in features
static constexpr int OO = 64;    // out features
static constexpr float LEAKY = 0.2f;

// ---------------------------------------------------------------------------
// Kernel 1: d[b][n] = rsqrt(1 + sum_m A[b][n][m])   (the +1 is the identity)
// One 256-thread block per row; LDS tree reduction.
// ---------------------------------------------------------------------------
__global__ __launch_bounds__(256)
void degree_kernel(const float* __restrict__ A, float* __restrict__ d) {
  const int row = blockIdx.x;                 // flat b*N + n, 0..B*N-1
  const float* Ar = A + (size_t)row * NN;
  float s = 0.f;
  for (int i = threadIdx.x; i < NN; i += 256) s += Ar[i];
  __shared__ float red[256];
  red[threadIdx.x] = s;
  __syncthreads();
  #pragma unroll
  for (int off = 128; off > 0; off >>= 1) {
    if (threadIdx.x < off) red[threadIdx.x] += red[threadIdx.x + off];
    __syncthreads();
  }
  if (threadIdx.x == 0) d[row] = rsqrtf(red[0] + 1.0f);
}

// ---------------------------------------------------------------------------
// Kernel 2: XWd[b][m][o] = d[b][m] * sum_f X[b][m][f] * W[f][o]
// One wave per 16-row tile; 4 column tiles (o) per wave; K=F=64 in steps of 4
// via V_WMMA_F32_16X16X4_F32.
// ---------------------------------------------------------------------------
__global__ __launch_bounds__(32)
void xw_kernel(const float* __restrict__ X, const float* __restrict__ W,
               const float* __restrict__ d, float* __restrict__ XWd) {
  const int tile = blockIdx.x;        // row tile 0..N/16-1
  const int b    = blockIdx.y;
  const int lane = threadIdx.x;
  const int l15  = lane & 15;
  const int half = lane >> 4;
  const int rowbase = tile * 16;

  const float* Xb = X + ((size_t)b * NN + rowbase) * FF;
  v8f acc[4] = {};

  for (int kb = 0; kb < FF; kb += 4) {
    const int k0 = kb + 2 * half;                     // this lane's K pair
    v2f a = *(const v2f*)(Xb + (size_t)l15 * FF + k0);
    #pragma unroll
    for (int c = 0; c < 4; ++c) {
      v2f bf;
      bf.x = W[(size_t)(k0    ) * OO + c * 16 + l15];
      bf.y = W[(size_t)(k0 + 1) * OO + c * 16 + l15];
      acc[c] = __builtin_amdgcn_wmma_f32_16x16x4_f32(
          false, a, false, bf, (short)0, acc[c], false, false);
    }
  }

  const float* db  = d + (size_t)b * NN;
  float*       out = XWd + ((size_t)b * NN + rowbase) * OO;
  #pragma unroll
  for (int r = 0; r < 8; ++r) {
    const int row = r + 8 * half;                     // row within tile
    const float s = db[rowbase + row];
    #pragma unroll
    for (int c = 0; c < 4; ++c)
      out[(size_t)row * OO + c * 16 + l15] = acc[c][r] * s;
  }
}

// ---------------------------------------------------------------------------
// Kernel 3: out[b][n][o] = leaky( d[n]*( sum_m A[b][n][m]*XWd[b][m][o]
//                                        + XWd[b][n][o] )  + bias[o] )
// One wave per 16-row tile computes all 64 output columns (4 accumulators),
// so A is streamed from HBM exactly once. K=N=2048 in steps of 4.
// ---------------------------------------------------------------------------
__global__ __launch_bounds__(32)
void gcn_kernel(const float* __restrict__ A, const float* __restrict__ XWd,
                const float* __restrict__ d, const float* __restrict__ bias,
                float* __restrict__ Xout) {
  const int tile = blockIdx.x;
  const int b    = blockIdx.y;
  const int lane = threadIdx.x;
  const int l15  = lane & 15;
  const int half = lane >> 4;
  const int rowbase = tile * 16;

  const float* Ab = A   + ((size_t)b * NN + rowbase) * NN;  // 16 rows of A
  const float* Wb = XWd + (size_t)b * NN * OO;              // L2-resident B
  v8f acc[4] = {};

  for (int kb = 0; kb < NN; kb += 4) {
    const int k0 = kb + 2 * half;
    v2f a = *(const v2f*)(Ab + (size_t)l15 * NN + k0);      // 8B load / lane
    #pragma unroll
    for (int c = 0; c < 4; ++c) {
      v2f bf;
      bf.x = Wb[(size_t)(k0    ) * OO + c * 16 + l15];
      bf.y = Wb[(size_t)(k0 + 1) * OO + c * 16 + l15];
      acc[c] = __builtin_amdgcn_wmma_f32_16x16x4_f32(
          false, a, false, bf, (short)0, acc[c], false, false);
    }
  }

  const float* db  = d + (size_t)b * NN;
  float*       out = Xout + ((size_t)b * NN + rowbase) * OO;
  #pragma unroll
  for (int r = 0; r < 8; ++r) {
    const int   row = r + 8 * half;
    const float dn  = db[rowbase + row];
    #pragma unroll
    for (int c = 0; c < 4; ++c) {
      const int col = c * 16 + l15;
      // fold the (A + I) diagonal term, then scale, bias, leaky-relu
      float v = acc[c][r] + Wb[(size_t)(rowbase + row) * OO + col];
      v = dn * v + bias[col];
      out[(size_t)row * OO + col] = (v >= 0.f) ? v : LEAKY * v;
    }
  }
}

// ---------------------------------------------------------------------------
extern "C" void kernel_launch(void* const* d_in, const int* in_sizes, int n_in,
                              void* d_out, int out_size, void* d_ws, size_t ws_size,
                              hipStream_t stream) {
  const float* X    = (const float*)d_in[0];   // [B,N,F]
  const float* A    = (const float*)d_in[1];   // [B,N,N]
  const float* W    = (const float*)d_in[2];   // [F,O]
  const float* bias = (const float*)d_in[3];   // [O]

  float* Xout = (float*)d_out;                              // [B,N,O] first
  float* Aout = Xout + (size_t)BB * NN * OO;                // then A verbatim

  float* dvec = (float*)d_ws;                               // B*N floats (64 KB)
  float* XWd  = dvec + (size_t)BB * NN;                     // B*N*O floats (4 MB)

  // 1) degrees
  degree_kernel<<<dim3(BB * NN), 256, 0, stream>>>(A, dvec);
  // 2) XWd = diag(d) * X * W
  xw_kernel<<<dim3(NN / 16, BB), 32, 0, stream>>>(X, W, dvec, XWd);
  // 3) main normalized-adjacency GEMM + epilogue
  gcn_kernel<<<dim3(NN / 16, BB), 32, 0, stream>>>(A, XWd, dvec, bias, Xout);
  // 4) pass-through second output: A
  hipMemcpyAsync(Aout, A, (size_t)BB * NN * NN * sizeof(float),
                 hipMemcpyDeviceToDevice, stream);
}